// Dynamic_Fusion_60249801228393
// MI455X (gfx1250) — compile-verified
//
#include <hip/hip_runtime.h>
#include <hip/hip_bf16.h>
#include <stddef.h>

// ---------------------------------------------------------------------------
// Shapes (fixed by the reference): B x N x Z points, B x (N-1) x Z vectors,
// B x H x N x N attention, N=17, Z=512, H=8.
// out[b,v,c] = Fa*(update[b,c] - drop[b,v,c]) + points[b,v,c]
//   drop  = PATH(17x16, 0/1 constant) @ vectors[b] (16x512)  -> fp32 WMMA
//   update= cand[b, argmax_v scores[b,v]]
//
// Roofline: ~457 MB traffic, ~1.2 GFLOP -> pure HBM-bound (~20us @ 23.3 TB/s).
// Dataset (457MB) > L2 (192MB) and is single-use -> non-temporal hints on the
// streaming accesses; WMMA f32 16x16x4 does the PATH einsum on the otherwise
// idle matrix pipe with exact fp32 math (PATH entries are 0/1).
// ---------------------------------------------------------------------------

#define NN   17
#define ZZ   512
#define HH   8

typedef __attribute__((ext_vector_type(2))) float v2f;
typedef __attribute__((ext_vector_type(8))) float v8f;

#if __has_builtin(__builtin_amdgcn_wmma_f32_16x16x4_f32)
#define HAVE_WMMA_F32X4 1
#endif

// PATH row bitmasks: bit e of mask[v] == PATH[v][e]  (e = node u-1 on path v->root)
__device__ __forceinline__ unsigned path_mask(int v) {
    const unsigned long long P0 = 0x0007000300010000ULL; // v=0..3
    const unsigned long long P1 = 0x0040003800180008ULL; // v=4..7
    const unsigned long long P2 = 0x04C003C001C000C0ULL; // v=8..11
    const unsigned long long P3 = 0x60C020C01CC00CC0ULL; // v=12..15
    if (v >= 16) return 0xE0C0u;
    unsigned long long pk = (v < 4) ? P0 : (v < 8) ? P1 : (v < 12) ? P2 : P3;
    return (unsigned)(pk >> ((v & 3) * 16)) & 0xFFFFu;
}

// ---------------------------------------------------------------------------
// Kernel 1: per-batch scores + stable argmax (top-1). One wave per batch.
// attn is read exactly once across the grid -> non-temporal loads.
// ---------------------------------------------------------------------------
__global__ __launch_bounds__(32) void topk_kernel(const float* __restrict__ attn,
                                                  int* __restrict__ top) {
    const int b    = blockIdx.x;
    const int lane = threadIdx.x;
    __shared__ float sc[NN];

    if (lane < NN) {
        const float* base = attn + (size_t)b * HH * NN * NN + (size_t)lane * NN;
        float s = 0.f;
#pragma unroll
        for (int h = 0; h < HH; ++h) {
            const float* row = base + (size_t)h * NN * NN;
#pragma unroll
            for (int j = 0; j < NN; ++j) s += __builtin_nontemporal_load(row + j);
        }
        sc[lane] = s;   // mean over h is a positive scale: argmax-invariant
    }
    __syncthreads();
    if (lane == 0) {
        int   best = 0;
        float bv   = sc[0];
#pragma unroll
        for (int v = 1; v < NN; ++v) {
            if (sc[v] > bv) { bv = sc[v]; best = v; }  // strict > == stable first-max
        }
        top[b] = best;
    }
}

// ---------------------------------------------------------------------------
// Kernel 2: fused drop (WMMA f32 16x16x4) + update + fma + store.
// One wave per (batch b, 16-channel tile). Layouts per CDNA5 ISA 7.12.2:
//   A 16x4 : lane<16 -> M=lane, K={4s,4s+1}; lane>=16 -> M=lane-16, K={4s+2,4s+3}
//   B 4x16 : vgpr j, lane<16 -> K=4s+j, N=lane; lane>=16 -> K=4s+j+2, N=lane-16
//   D 16x16: vgpr r, lane<16 -> M=r,    N=lane; lane>=16 -> M=r+8,   N=lane-16
// ---------------------------------------------------------------------------
__global__ __launch_bounds__(256) void fuse_kernel(const float* __restrict__ points,
                                                   const float* __restrict__ vectors,
                                                   const float* __restrict__ Fa_p,
                                                   const int*   __restrict__ top_p,
                                                   float* __restrict__ out,
                                                   int totalWaves) {
    const int lane  = threadIdx.x & 31;
    const int w     = (int)(blockIdx.x * (blockDim.x >> 5)) + (threadIdx.x >> 5);
    if (w >= totalWaves) return;                 // wave-uniform: EXEC stays all-1s

    const int b    = w >> 5;                     // 32 channel tiles per batch
    const int c0   = (w & 31) << 4;              // tile channel base
    const int m    = lane & 15;                  // N column / M row within half
    const int half = lane >> 4;                  // 0: lanes 0-15, 1: lanes 16-31
    const int koff = half << 1;

    const size_t pbase = (size_t)b * NN * ZZ;
    const size_t vbase = (size_t)b * (NN - 1) * ZZ;
    const int    ch    = c0 + m;

    // ---- load B tiles (vectors 16x16) in WMMA layout: 8 x b32 NT loads ----
    v2f Bv[4];
#pragma unroll
    for (int s = 0; s < 4; ++s) {
        const int k0 = 4 * s + koff;
        Bv[s].x = __builtin_nontemporal_load(vectors + vbase + (size_t)(k0 + 0) * ZZ + ch);
        Bv[s].y = __builtin_nontemporal_load(vectors + vbase + (size_t)(k0 + 1) * ZZ + ch);
    }

    float dropv[8];
    float drop16;

#if defined(HAVE_WMMA_F32X4)
    // ---- A from constant PATH bitmasks; chain 4 WMMAs over K=16 ----
    const unsigned mask   = path_mask(m);
    const unsigned mask16 = 0xE0C0u;             // PATH row v=16
    v8f D   = {};
    v8f D16 = {};
#pragma unroll
    for (int s = 0; s < 4; ++s) {
        v2f A, A2;
        A.x = (float)((mask >> (4 * s + koff + 0)) & 1u);
        A.y = (float)((mask >> (4 * s + koff + 1)) & 1u);
        // replicate PATH[16] into rows 0 AND 8 so both lane halves get drop16
        const float r0 = (float)((mask16 >> (4 * s + koff + 0)) & 1u);
        const float r1 = (float)((mask16 >> (4 * s + koff + 1)) & 1u);
        const bool  on = (m == 0) || (m == 8);
        A2.x = on ? r0 : 0.f;
        A2.y = on ? r1 : 0.f;
        D   = __builtin_amdgcn_wmma_f32_16x16x4_f32(false, A,  false, Bv[s],
                                                    (short)0, D,   false, false);
        D16 = __builtin_amdgcn_wmma_f32_16x16x4_f32(false, A2, false, Bv[s],
                                                    (short)0, D16, false, false);
    }
#pragma unroll
    for (int r = 0; r < 8; ++r) dropv[r] = D[r];
    drop16 = D16[0];
#else
    // ---- scalar fallback: per-lane path sums over directly loaded vectors ----
    float vec[16];
#pragma unroll
    for (int e = 0; e < 16; ++e)
        vec[e] = __builtin_nontemporal_load(vectors + vbase + (size_t)e * ZZ + ch);
#pragma unroll
    for (int r = 0; r < 8; ++r) {
        const unsigned mk = path_mask(r + half * 8);
        float s = 0.f;
#pragma unroll
        for (int e = 0; e < 16; ++e)
            if ((mk >> e) & 1u) s += vec[e];
        dropv[r] = s;
    }
    {
        const unsigned mk = 0xE0C0u;
        float s = 0.f;
#pragma unroll
        for (int e = 0; e < 16; ++e)
            if ((mk >> e) & 1u) s += vec[e];
        drop16 = s;
    }
#endif

    // ---- update[b, ch] from top-1 (has reuse: keep cached/regular loads) ----
    const float Fa  = Fa_p[0];
    const int   top = top_p[b];
    float upd;
    if (top == 0) {
        upd = points[pbase + ch];
    } else {
        const int a = (top < 4) ? 1 : (top < 7) ? 4 : 7;     // ANC1[top]
        upd = points[pbase + (size_t)a * ZZ + ch] +
              vectors[vbase + (size_t)(a - 1) * ZZ + ch];
    }

    // ---- fused output: rows 0..15 via D layout, row 16 from lanes 0-15 ----
#pragma unroll
    for (int r = 0; r < 8; ++r) {
        const int    v   = r + half * 8;
        const size_t off = pbase + (size_t)v * ZZ + ch;
        const float  p   = __builtin_nontemporal_load(points + off);
        __builtin_nontemporal_store(fmaf(Fa, upd - dropv[r], p), out + off);
    }
    if (half == 0) {
        const size_t off = pbase + (size_t)16 * ZZ + ch;
        const float  p   = __builtin_nontemporal_load(points + off);
        __builtin_nontemporal_store(fmaf(Fa, upd - drop16, p), out + off);
    }
}

// ---------------------------------------------------------------------------
extern "C" void kernel_launch(void* const* d_in, const int* in_sizes, int n_in,
                              void* d_out, int out_size, void* d_ws, size_t ws_size,
                              hipStream_t stream) {
    const float* points  = (const float*)d_in[0];
    const float* vectors = (const float*)d_in[1];
    const float* attn    = (const float*)d_in[2];
    const float* Fa      = (const float*)d_in[3];
    float*       out     = (float*)d_out;
    int*         top     = (int*)d_ws;                 // B ints of scratch

    const int B = in_sizes[0] / (NN * ZZ);             // 4096

    topk_kernel<<<B, 32, 0, stream>>>(attn, top);

    const int totalWaves    = B * (ZZ / 16);           // 131072
    const int wavesPerBlock = 8;                       // 256 threads
    const int blocks        = (totalWaves + wavesPerBlock - 1) / wavesPerBlock;
    fuse_kernel<<<blocks, 256, 0, stream>>>(points, vectors, Fa, top, out, totalWaves);
}